// SparseGraphTransformerAttention_54932631716252
// MI455X (gfx1250) — compile-verified
//
#include <hip/hip_runtime.h>
#include <hip/hip_bf16.h>
#include <math.h>

#define N_NODES  50000
#define N_EDGES  1600000
#define IN_F     128
#define D_ATT    64
#define INV_SQRT_DHEAD 0.35355339059327373f
#define EPS_SM   1e-16f

typedef __attribute__((ext_vector_type(2))) float v2f;
typedef __attribute__((ext_vector_type(8))) float v8f;

// ---------------------------------------------------------------------------
// Projection: out(N,64) = x(N,128) @ W(64,128)^T + b, via V_WMMA_F32_16X16X4_F32.
// One wave -> one 16-row tile, 4 column tiles (64 cols), K=128 in steps of 4.
// permute!=0 stores v in (N, d_head, n_heads) order: out[n*64 + (c&7)*8 + (c>>3)].
// ---------------------------------------------------------------------------
__global__ __launch_bounds__(256) void proj_wmma(
    const float* __restrict__ x, const float* __restrict__ W,
    const float* __restrict__ bias, float* __restrict__ out, int permute)
{
    const int lane  = threadIdx.x & 31;
    const int wave  = threadIdx.x >> 5;
    const int tile  = blockIdx.x * 8 + wave;
    if (tile * 16 >= N_NODES) return;          // wave-uniform exit -> EXEC all-1s below

    const int l16   = lane & 15;
    const int khalf = (lane >> 4) << 1;        // lanes 0-15: K+0/1, lanes 16-31: K+2/3
    const int row   = tile * 16 + l16;

    const float2* __restrict__ xr = (const float2*)(x + (size_t)row * IN_F);
    const float2* __restrict__ w0 = (const float2*)(W + (size_t)( 0 + l16) * IN_F);
    const float2* __restrict__ w1 = (const float2*)(W + (size_t)(16 + l16) * IN_F);
    const float2* __restrict__ w2 = (const float2*)(W + (size_t)(32 + l16) * IN_F);
    const float2* __restrict__ w3 = (const float2*)(W + (size_t)(48 + l16) * IN_F);

    v8f acc0 = {}, acc1 = {}, acc2 = {}, acc3 = {};

    #pragma unroll 4
    for (int kb = 0; kb < IN_F; kb += 4) {
        const int ki = (kb + khalf) >> 1;      // float2 index

        float2 av = xr[ki];
        v2f a; a.x = av.x; a.y = av.y;

        float2 b0v = w0[ki]; v2f b0; b0.x = b0v.x; b0.y = b0v.y;
        float2 b1v = w1[ki]; v2f b1; b1.x = b1v.x; b1.y = b1v.y;
        float2 b2v = w2[ki]; v2f b2; b2.x = b2v.x; b2.y = b2v.y;
        float2 b3v = w3[ki]; v2f b3; b3.x = b3v.x; b3.y = b3v.y;

        acc0 = __builtin_amdgcn_wmma_f32_16x16x4_f32(false, a, false, b0, (short)0, acc0, false, false);
        acc1 = __builtin_amdgcn_wmma_f32_16x16x4_f32(false, a, false, b1, (short)0, acc1, false, false);
        acc2 = __builtin_amdgcn_wmma_f32_16x16x4_f32(false, a, false, b2, (short)0, acc2, false, false);
        acc3 = __builtin_amdgcn_wmma_f32_16x16x4_f32(false, a, false, b3, (short)0, acc3, false, false);
    }

    // C/D layout: VGPR r -> (M = r + 8*(lane>=16), N = lane%16) within the 16x16 tile.
    const int mrow_base = tile * 16 + ((lane >> 4) << 3);
    #pragma unroll
    for (int r = 0; r < 8; ++r) {
        const size_t m = (size_t)(mrow_base + r);
        #pragma unroll
        for (int c = 0; c < 4; ++c) {
            const int n = c * 16 + l16;
            float val = (c == 0 ? acc0[r] : c == 1 ? acc1[r] : c == 2 ? acc2[r] : acc3[r])
                        + bias[n];
            if (permute)
                out[m * D_ATT + (n & 7) * 8 + (n >> 3)] = val;  // (N, d_head, n_heads)
            else
                out[m * D_ATT + n] = val;                       // linear (N, 64)
        }
    }
}

// ---------------------------------------------------------------------------
// Init segment-max to -inf, segment-sum to 0.
// ---------------------------------------------------------------------------
__global__ __launch_bounds__(256) void init_ms(float* __restrict__ m, float* __restrict__ s)
{
    int i = blockIdx.x * blockDim.x + threadIdx.x;
    if (i < N_NODES * 8) { m[i] = -INFINITY; s[i] = 0.0f; }
}

__device__ __forceinline__ void atomicMaxF32(float* addr, float val)
{
    // Monotonic-bit-pattern trick: int max for >=0, uint min for <0. Init = -inf.
    if (val >= 0.0f) atomicMax((int*)addr, __float_as_int(val));
    else             atomicMin((unsigned int*)addr, __float_as_uint(val));
}

// ---------------------------------------------------------------------------
// Pass 1: per (edge, head) logit = (q[src]·k[dst]) / sqrt(8) * edge_attr;
// store prods, atomic segment-max keyed by dst. q/k gathers hit L2 (25.6MB << 192MB).
// ---------------------------------------------------------------------------
__global__ __launch_bounds__(256) void edge_logits(
    const float* __restrict__ q, const float* __restrict__ k,
    const long long* __restrict__ edge, const float* __restrict__ eattr,
    float* __restrict__ prods_out, float* __restrict__ mmax)
{
    int gid = blockIdx.x * blockDim.x + threadIdx.x;
    if (gid >= N_EDGES * 8) return;
    const int e = gid >> 3, h = gid & 7;
    const int s = (int)edge[e];
    const int d = (int)edge[N_EDGES + e];

    const float4* qp = (const float4*)(q + (size_t)s * D_ATT + h * 8);
    const float4* kp = (const float4*)(k + (size_t)d * D_ATT + h * 8);
    float4 q0 = qp[0], q1 = qp[1];
    float4 k0 = kp[0], k1 = kp[1];
    float dot = q0.x*k0.x + q0.y*k0.y + q0.z*k0.z + q0.w*k0.w
              + q1.x*k1.x + q1.y*k1.y + q1.z*k1.z + q1.w*k1.w;

    float logit = dot * INV_SQRT_DHEAD * eattr[e];
    prods_out[gid] = logit;
    atomicMaxF32(mmax + (size_t)d * 8 + h, logit);
}

// ---------------------------------------------------------------------------
// Pass 2: ex = exp(logit - max[dst]); stash ex in attention slot; atomic segment-sum.
// ---------------------------------------------------------------------------
__global__ __launch_bounds__(256) void edge_exp(
    const float* __restrict__ prods, const long long* __restrict__ edge,
    const float* __restrict__ mmax, float* __restrict__ att, float* __restrict__ ssum)
{
    int gid = blockIdx.x * blockDim.x + threadIdx.x;
    if (gid >= N_EDGES * 8) return;
    const int e = gid >> 3, h = gid & 7;
    const int d = (int)edge[N_EDGES + e];
    float ex = expf(prods[gid] - mmax[(size_t)d * 8 + h]);
    att[gid] = ex;
    atomicAdd(ssum + (size_t)d * 8 + h, ex);
}

// ---------------------------------------------------------------------------
// Pass 3: attention = ex / (sum[dst] + eps).
// ---------------------------------------------------------------------------
__global__ __launch_bounds__(256) void edge_norm(
    float* __restrict__ att, const long long* __restrict__ edge,
    const float* __restrict__ ssum)
{
    int gid = blockIdx.x * blockDim.x + threadIdx.x;
    if (gid >= N_EDGES * 8) return;
    const int e = gid >> 3, h = gid & 7;
    const int d = (int)edge[N_EDGES + e];
    att[gid] = att[gid] / (ssum[(size_t)d * 8 + h] + EPS_SM);
}

extern "C" void kernel_launch(void* const* d_in, const int* in_sizes, int n_in,
                              void* d_out, int out_size, void* d_ws, size_t ws_size,
                              hipStream_t stream)
{
    const float*     x     = (const float*)d_in[0];
    const float*     Wq    = (const float*)d_in[1];
    const float*     bq    = (const float*)d_in[2];
    const float*     Wk    = (const float*)d_in[3];
    const float*     bk    = (const float*)d_in[4];
    const float*     Wv    = (const float*)d_in[5];
    const float*     bv    = (const float*)d_in[6];
    const float*     eattr = (const float*)d_in[7];
    const long long* edge  = (const long long*)d_in[8];   // int64 (2, E)

    float* out   = (float*)d_out;
    float* att   = out;                                   // (E, 8)
    float* vout  = att  + (size_t)N_EDGES * 8;            // (N, d_head, n_heads)
    float* prods = vout + (size_t)N_NODES * D_ATT;        // (E, 8)

    float* q    = (float*)d_ws;                           // (N, 64) head-major h*8+d
    float* kbuf = q    + (size_t)N_NODES * D_ATT;
    float* mmax = kbuf + (size_t)N_NODES * D_ATT;         // (N, 8)
    float* ssum = mmax + (size_t)N_NODES * 8;             // (N, 8)

    const int tiles   = (N_NODES + 15) / 16;              // 3125
    const int pblocks = (tiles + 7) / 8;                  // 8 waves/block
    proj_wmma<<<pblocks, 256, 0, stream>>>(x, Wq, bq, q,    0);
    proj_wmma<<<pblocks, 256, 0, stream>>>(x, Wk, bk, kbuf, 0);
    proj_wmma<<<pblocks, 256, 0, stream>>>(x, Wv, bv, vout, 1);

    init_ms<<<(N_NODES * 8 + 255) / 256, 256, 0, stream>>>(mmax, ssum);

    const int eth     = N_EDGES * 8;
    const int eblocks = (eth + 255) / 256;
    edge_logits<<<eblocks, 256, 0, stream>>>(q, kbuf, edge, eattr, prods, mmax);
    edge_exp   <<<eblocks, 256, 0, stream>>>(prods, edge, mmax, att, ssum);
    edge_norm  <<<eblocks, 256, 0, stream>>>(att, edge, ssum);
}